// RotRCNN_15358803050973
// MI455X (gfx1250) — compile-verified
//
#include <hip/hip_runtime.h>

// CDNA5 / gfx1250, wave32. All GEMM-shaped work on V_WMMA_F32_16X16X4_F32
// (f32 reference, ~15 GFLOP / ~0.5 GB => memory-bound at 23.3 TB/s, so the
// f32 matrix path is the right precision: exact numerics, matrix pipe
// engaged). Convs: implicit GEMM, tap-major K, address-clamped edges (EXEC
// stays full), transposed weights for coalesced B loads. FC: block-level
// tiling with a double-buffered LDS weight tile shared by 8 waves
// (ds_load/ds_store + workgroup barrier), software-pipelined against WMMA.

typedef __attribute__((ext_vector_type(2))) float v2f;
typedef __attribute__((ext_vector_type(8))) float v8f;

#define LEAKY 0.2f

// ---------------------------------------------------------------- zero fill
__global__ void rr_zero_kernel(float* __restrict__ p, int n) {
    int i = blockIdx.x * blockDim.x + threadIdx.x;
    int stride = gridDim.x * blockDim.x;
    for (; i < n; i += stride) p[i] = 0.0f;
}

// -------------------------------------------- rotate + project + scatter
__global__ void rr_rot_scatter_kernel(const float* __restrict__ pts,
                                      const float* __restrict__ rot,
                                      float* __restrict__ img,
                                      int B, int N) {
    int idx = blockIdx.x * blockDim.x + threadIdx.x;
    if (idx >= B * N) return;
    int b = idx / N;
    float x = pts[(long)idx * 3 + 0];
    float y = pts[(long)idx * 3 + 1];
    float z = pts[(long)idx * 3 + 2];
    #pragma unroll 1
    for (int m = 0; m < 12; ++m) {
        const float* R = rot + m * 9;
        float xr = R[0] * x + R[1] * y + R[2] * z;
        float yr = R[3] * x + R[4] * y + R[5] * z;
        float zr = R[6] * x + R[7] * y + R[8] * z;
        int px = (int)rintf((xr + 2.0f) * 16.0f);   // /0.0625
        int py = (int)rintf((yr + 2.0f) * 16.0f);
        bool ok = (px >= 0) && (px < 64) && (py >= 0) && (py < 64);
        if (!ok) { px = 0; py = 0; }
        img[((((long)b * 12 + m) * 64) + py) * 64 + px] = zr * 0.1f;
    }
}

// --------------------------------------- weight transpose: OIHW -> [9][IC][OC]
__global__ void rr_wt_kernel(const float* __restrict__ w, float* __restrict__ wT,
                             int OC, int IC) {
    int i = blockIdx.x * blockDim.x + threadIdx.x;
    int total = OC * IC * 9;
    if (i >= total) return;
    int oc = i / (IC * 9);
    int rem = i % (IC * 9);
    int ic = rem / 9;
    int t  = rem % 9;
    wT[(t * IC + ic) * OC + oc] = w[i];
}

// ------------------------------------------ stride-2 3x3 conv, implicit GEMM
// K ordering: k = tap*IC + ic (tap loop fully unrolled, constant ky/kx).
// WMMA row m only affects output row m => clamp addresses for edge rows and
// never store them; EXEC stays all-ones throughout.
template<int IC, int IH, int OC, int OH>
__global__ void rr_conv_s2_wmma(const float* __restrict__ in,
                                const float* __restrict__ wT,   // [9][IC][OC]
                                const float* __restrict__ bias,
                                float* __restrict__ out,
                                int totalWaves) {
    constexpr int IW = IH, OW = OH;
    constexpr int tilesN = OC / 16;
    constexpr int tilesM = (OH * OW + 15) / 16;
    constexpr int CH = IH * IW;

    int lane = threadIdx.x & 31;
    int gw = blockIdx.x * (blockDim.x >> 5) + (threadIdx.x >> 5);
    if (gw >= totalWaves) return;               // wave-uniform

    int b  = gw / (tilesM * tilesN);
    int r2 = gw % (tilesM * tilesN);
    int tm = r2 / tilesN, tn = r2 % tilesN;

    int mGlob  = tm * 16 + (lane & 15);
    int mClamp = (mGlob < OH * OW) ? mGlob : (OH * OW - 1);
    int oy = mClamp / OW, ox = mClamp % OW;
    int nGlob = tn * 16 + (lane & 15);          // OC multiple of 16
    int khalf = (lane >> 4) & 1;

    const float* inB = in + (long)b * IC * CH + (oy * 2) * IW + (ox * 2)
                          + (long)khalf * 2 * CH;
    const float* wB  = wT + (long)khalf * 2 * OC + nGlob;

    v8f acc = {};
    #pragma unroll
    for (int t = 0; t < 9; ++t) {
        const int ky = t / 3, kx = t % 3;       // compile-time constants
        const float* aP = inB + ky * IW + kx;
        const float* bP = wB + (long)t * IC * OC;
        #pragma unroll 2
        for (int ic = 0; ic < IC; ic += 4) {
            v2f aF, bF;
            aF.x = aP[0];  aF.y = aP[CH];
            bF.x = bP[0];  bF.y = bP[OC];
            aP += 4 * CH;  bP += 4 * OC;
            acc = __builtin_amdgcn_wmma_f32_16x16x4_f32(
                      false, aF, false, bF, (short)0, acc, false, false);
        }
    }

    float bn = bias[nGlob];
    #pragma unroll
    for (int r = 0; r < 8; ++r) {
        int mrow = tm * 16 + r + ((lane & 16) ? 8 : 0);
        if (mrow < OH * OW) {
            float v = acc[r] + bn;
            v = (v >= 0.0f) ? v : LEAKY * v;
            out[((long)b * OC + nGlob) * (OH * OW) + mrow] = v;
        }
    }
}

// ---------------------------------------------------------- FC: out = in*W^T+b
// Block = 8 waves = 8 M-tiles x 1 N-tile. The 16-column weight tile is shared:
// staged cooperatively into LDS in 64-deep K-chunks, double-buffered so the
// next chunk's global->LDS copy overlaps WMMA on the current chunk.
// LDS layout [k][16]: kk*16 = 0 mod 64 => fragment reads are bank-conflict-free
// (lanes 0-15 banks n / 16+n, lanes 16-31 banks 32+n / 48+n).
template<int K>
__global__ void rr_fc_wmma(const float* __restrict__ in,   // M x K
                           const float* __restrict__ w,    // N x K
                           const float* __restrict__ bias,
                           float* __restrict__ out,        // M x N
                           int N, int doRelu) {
    constexpr int KC = 64;                      // K-chunk in LDS
    __shared__ float bS[2][KC * 16];            // 8 KB double-buffered B tile

    int tid  = threadIdx.x;
    int lane = tid & 31;
    int wv   = tid >> 5;                        // 0..7
    int tilesN = (N + 15) >> 4;
    int tn = blockIdx.x % tilesN;
    int tm = (blockIdx.x / tilesN) * 8 + wv;    // tilesM = 32, divisible by 8

    int mGlob  = tm * 16 + (lane & 15);         // M = 512, always valid
    int nGlob  = tn * 16 + (lane & 15);
    int khalf  = (lane >> 4) & 1;
    const float* aRow = in + (long)mGlob * K + khalf * 2;

    // Cooperative staging map: 256 threads, 1024 elems/chunk, 4 per thread.
    // k = tid&63 (lanes coalesced along K), n = tid>>6 (+4 per pass).
    int sk  = tid & 63;
    int sn0 = tid >> 6;                         // 0..3
    long bOff[4];
    #pragma unroll
    for (int p = 0; p < 4; ++p) {
        int n  = sn0 + p * 4;
        int nc = (tn * 16 + n < N) ? (tn * 16 + n) : (N - 1);   // clamp tail cols
        bOff[p] = (long)nc * K + sk;
    }

    // Prologue: stage chunk 0.
    #pragma unroll
    for (int p = 0; p < 4; ++p)
        bS[0][sk * 16 + (sn0 + p * 4)] = w[bOff[p]];
    __syncthreads();

    v8f acc = {};
    int buf = 0;
    for (int kc = 0; kc < K; kc += KC) {
        // Stage next chunk into the other buffer (overlaps WMMA below).
        if (kc + KC < K) {
            #pragma unroll
            for (int p = 0; p < 4; ++p)
                bS[buf ^ 1][sk * 16 + (sn0 + p * 4)] = w[bOff[p] + kc + KC];
        }
        if (kc + 256 < K)                       // uniform GL2 prefetch of A stream
            __builtin_prefetch(aRow + kc + 256, 0, 0);

        const float* aP = aRow + kc;
        const float* bL = &bS[buf][khalf * 2 * 16 + (lane & 15)];
        #pragma unroll 4
        for (int kk = 0; kk < KC; kk += 4) {
            v2f aF, bF;
            aF.x = aP[kk];           aF.y = aP[kk + 1];
            bF.x = bL[kk * 16];      bF.y = bL[kk * 16 + 16];
            acc = __builtin_amdgcn_wmma_f32_16x16x4_f32(
                      false, aF, false, bF, (short)0, acc, false, false);
        }
        __syncthreads();                        // staging done & reads done
        buf ^= 1;
    }

    int nClamp = (nGlob < N) ? nGlob : (N - 1);
    float bn = bias[nClamp];
    #pragma unroll
    for (int r = 0; r < 8; ++r) {
        int mrow = tm * 16 + r + ((lane & 16) ? 8 : 0);
        if (nGlob < N) {
            float v = acc[r] + bn;
            if (doRelu) v = fmaxf(v, 0.0f);
            out[(long)mrow * N + nGlob] = v;
        }
    }
}

// ---------------------------------------------------------------- launcher
template<int K>
static inline void launch_fc(const float* in, const float* w, const float* b,
                             float* out, int M, int N, int relu,
                             hipStream_t stream) {
    int blocks = ((M >> 4) / 8) * ((N + 15) >> 4);   // exact: no partial blocks
    rr_fc_wmma<K><<<blocks, 256, 0, stream>>>(in, w, b, out, N, relu);
}

extern "C" void kernel_launch(void* const* d_in, const int* in_sizes, int n_in,
                              void* d_out, int out_size, void* d_ws, size_t ws_size,
                              hipStream_t stream) {
    const float* pts   = (const float*)d_in[0];
    const float* rot   = (const float*)d_in[1];
    const float* cw1   = (const float*)d_in[2];
    const float* cb1   = (const float*)d_in[3];
    const float* cw2   = (const float*)d_in[4];
    const float* cb2   = (const float*)d_in[5];
    const float* cw3   = (const float*)d_in[6];
    const float* cb3   = (const float*)d_in[7];
    const float* clsw1 = (const float*)d_in[8];
    const float* clsb1 = (const float*)d_in[9];
    const float* clsw2 = (const float*)d_in[10];
    const float* clsb2 = (const float*)d_in[11];
    const float* clsw3 = (const float*)d_in[12];
    const float* clsb3 = (const float*)d_in[13];
    const float* regw1 = (const float*)d_in[14];
    const float* regb1 = (const float*)d_in[15];
    const float* regw2 = (const float*)d_in[16];
    const float* regb2 = (const float*)d_in[17];
    const float* regw3 = (const float*)d_in[18];
    const float* regb3 = (const float*)d_in[19];

    float* out = (float*)d_out;                 // [512*1 | 512*46] concatenated
    float* ws  = (float*)d_ws;

    const int B = 512, N = 512;
    // Workspace layout (floats), with reuse (peak ~164 MB):
    const long IMG_N = (long)B * 12 * 64 * 64;  // 25,165,824
    float* img = ws;                            // [0, 25.2M)
    float* c1  = ws + 25165824;                 // 15,745,024 (512*32*31*31)
    float* c2  = ws;                            // reuse img region: 7,372,800
    float* c3  = ws + 7372800;                  // 3,211,264 (= feat, 512x6272)
    float* h1  = ws + 10584064;                 // 131,072 (512x256)
    float* h2  = ws + 10715136;                 // 131,072
    float* wTr = ws + 40910848;                 // 73,728 max (reused per conv)

    // 1) images = 0, then scatter rotated projections
    rr_zero_kernel<<<2048, 256, 0, stream>>>(img, (int)IMG_N);
    rr_rot_scatter_kernel<<<(B * N + 255) / 256, 256, 0, stream>>>(pts, rot, img, B, N);

    // 2) convs (implicit-GEMM f32 WMMA; wTr reused — stream order serializes)
    {   // conv1: (12,64,64) -> (32,31,31)
        constexpr int IC = 12, IH = 64, OC = 32, OH = 31;
        rr_wt_kernel<<<(OC * IC * 9 + 255) / 256, 256, 0, stream>>>(cw1, wTr, OC, IC);
        int tw = B * (((OH * OH + 15) / 16) * (OC / 16));
        rr_conv_s2_wmma<IC, IH, OC, OH><<<(tw + 7) / 8, 256, 0, stream>>>(img, wTr, cb1, c1, tw);
    }
    {   // conv2: (32,31,31) -> (64,15,15)
        constexpr int IC = 32, IH = 31, OC = 64, OH = 15;
        rr_wt_kernel<<<(OC * IC * 9 + 255) / 256, 256, 0, stream>>>(cw2, wTr, OC, IC);
        int tw = B * (((OH * OH + 15) / 16) * (OC / 16));
        rr_conv_s2_wmma<IC, IH, OC, OH><<<(tw + 7) / 8, 256, 0, stream>>>(c1, wTr, cb2, c2, tw);
    }
    {   // conv3: (64,15,15) -> (128,7,7) == feat (512 x 6272)
        constexpr int IC = 64, IH = 15, OC = 128, OH = 7;
        rr_wt_kernel<<<(OC * IC * 9 + 255) / 256, 256, 0, stream>>>(cw3, wTr, OC, IC);
        int tw = B * (((OH * OH + 15) / 16) * (OC / 16));
        rr_conv_s2_wmma<IC, IH, OC, OH><<<(tw + 7) / 8, 256, 0, stream>>>(c2, wTr, cb3, c3, tw);
    }

    // 3) heads (LDS-staged, pipelined f32 WMMA GEMMs)
    launch_fc<6272>(c3, clsw1, clsb1, h1, 512, 256, 1, stream);
    launch_fc<256>(h1, clsw2, clsb2, h2, 512, 256, 1, stream);
    launch_fc<256>(h2, clsw3, clsb3, out, 512, 1, 0, stream);          // rcnn_cls

    launch_fc<6272>(c3, regw1, regb1, h1, 512, 256, 1, stream);
    launch_fc<256>(h1, regw2, regb2, h2, 512, 256, 1, stream);
    launch_fc<256>(h2, regw3, regb3, out + 512, 512, 46, 0, stream);   // rcnn_reg
}